// TopK_87909390614842
// MI455X (gfx1250) — compile-verified
//
#include <hip/hip_runtime.h>
#include <hip/hip_bf16.h>

typedef unsigned int uint32;
typedef unsigned long long u64;
typedef float v2f __attribute__((ext_vector_type(2)));
typedef float v8f __attribute__((ext_vector_type(8)));

#define FDIM 256
#define KSEL 4096
#define CAND_CAP 8192

struct TopKState {
  uint32 prefix;
  uint32 k_rem;
};

// Monotone map: float -> uint32 preserving order (larger float -> larger uint)
__device__ __forceinline__ uint32 f2ord(float f) {
  uint32 u = __float_as_uint(f);
  return (u & 0x80000000u) ? ~u : (u | 0x80000000u);
}

// ---------------------------------------------------------------------------
// Phase 0: normalize scorer: sU = scorer / (||scorer|| + 1e-8)
// ---------------------------------------------------------------------------
__global__ void prep_scorer_kernel(const float* __restrict__ scorer,
                                   float* __restrict__ sU_g) {
  __shared__ float red[FDIM];
  __shared__ float inv;
  int t = threadIdx.x;
  float v = scorer[t];
  red[t] = v * v;
  __syncthreads();
  for (int off = FDIM >> 1; off > 0; off >>= 1) {
    if (t < off) red[t] += red[t + off];
    __syncthreads();
  }
  if (t == 0) inv = 1.0f / (sqrtf(red[0]) + 1e-8f);
  __syncthreads();
  sU_g[t] = v * inv;
}

// ---------------------------------------------------------------------------
// Phase 1: scores[n] = dot(node_embs[n,:], sU) + mask[n]  via WMMA f32 16x16x4
// One wave computes 16 rows. A: 16x4 f32 tile of node_embs (lane L holds
// row L&15, K-half (L>>4)*2). B: 4x16 = sU chunk broadcast over N columns.
// D accumulates 16x16; every column holds the 16 scores.
// ---------------------------------------------------------------------------
__global__ void score_wmma_kernel(const float* __restrict__ embs,
                                  const float* __restrict__ sU_g,
                                  const float* __restrict__ mask,
                                  float* __restrict__ scores, int N) {
  __shared__ float sU[FDIM];
  sU[threadIdx.x] = sU_g[threadIdx.x];   // blockDim.x == 256
  __syncthreads();

  int wave = threadIdx.x >> 5;
  int lane = threadIdx.x & 31;
  int rowbase = (blockIdx.x * 8 + wave) * 16;

  int m = lane & 15;
  int khalf = (lane >> 4) * 2;          // 0 for lanes 0-15, 2 for lanes 16-31
  int lrow = rowbase + m;
  if (lrow >= N) lrow = N - 1;          // clamp loads; stores are guarded
  const float* p = embs + (size_t)lrow * FDIM + khalf;

  v8f c = {};
#pragma unroll 4
  for (int kb = 0; kb < FDIM; kb += 4) {
    const float2 av = *(const float2*)(p + kb);
    v2f a; a.x = av.x; a.y = av.y;
    v2f b; b.x = sU[kb + khalf]; b.y = sU[kb + khalf + 1];
    c = __builtin_amdgcn_wmma_f32_16x16x4_f32(false, a, false, b,
                                              (short)0, c, false, false);
  }

  // D layout: VGPR r -> M=r (lanes 0-15), M=r+8 (lanes 16-31)
  if ((lane & 15) == 0) {
    int half = lane >> 4;
#pragma unroll
    for (int r = 0; r < 8; ++r) {
      int row = rowbase + half * 8 + r;
      if (row < N) scores[row] = c[r] + mask[row];
    }
  }
}

// ---------------------------------------------------------------------------
// Top-k: 3-level radix select (12 + 12 + 8 bits)
// ---------------------------------------------------------------------------
__global__ void init_misc_kernel(TopKState* st, uint32* cnt,
                                 u64* __restrict__ cands) {
  int i = blockIdx.x * blockDim.x + threadIdx.x;
  if (i < CAND_CAP) cands[i] = 0ull;
  if (i == 0) { st->prefix = 0u; st->k_rem = KSEL; *cnt = 0u; }
}

__global__ void zero_hist_kernel(uint32* __restrict__ hist) {
  hist[blockIdx.x * blockDim.x + threadIdx.x] = 0u;
}

__global__ void hist_pass_kernel(const float* __restrict__ scores,
                                 uint32* __restrict__ hist,
                                 const TopKState* __restrict__ st,
                                 int N, int level) {
  __shared__ uint32 lh[4096];
  for (int i = threadIdx.x; i < 4096; i += blockDim.x) lh[i] = 0u;
  __syncthreads();
  uint32 pfx = (level > 0) ? st->prefix : 0u;
  int stride = gridDim.x * blockDim.x;
  for (int n = blockIdx.x * blockDim.x + threadIdx.x; n < N; n += stride) {
    uint32 s = f2ord(scores[n]);
    if (level == 0) {
      atomicAdd(&lh[s >> 20], 1u);
    } else if (level == 1) {
      if ((s >> 20) == pfx) atomicAdd(&lh[(s >> 8) & 0xFFFu], 1u);
    } else {
      if ((s >> 8) == pfx) atomicAdd(&lh[s & 0xFFu], 1u);
    }
  }
  __syncthreads();
  for (int i = threadIdx.x; i < 4096; i += blockDim.x)
    if (lh[i]) atomicAdd(&hist[i], lh[i]);
}

// Descending suffix scan over NB buckets; finds bucket where cumulative count
// (from highest value down) first reaches k_rem. Single block of 256 threads.
__global__ void scan_pass_kernel(const uint32* __restrict__ hist,
                                 TopKState* __restrict__ st,
                                 int NB, int bits) {
  __shared__ uint32 ps[256];
  __shared__ uint32 sh_krem, sh_prefix;
  int t = threadIdx.x;
  if (t == 0) { sh_krem = st->k_rem; sh_prefix = st->prefix; }
  __syncthreads();
  int CH = NB >> 8;  // 16 for NB=4096, 1 for NB=256
  uint32 local = 0;
  for (int i = 0; i < CH; ++i) local += hist[NB - 1 - (t * CH + i)];
  ps[t] = local;
  __syncthreads();
  for (int off = 1; off < 256; off <<= 1) {
    uint32 v = (t >= off) ? ps[t - off] : 0u;
    __syncthreads();
    ps[t] += v;
    __syncthreads();
  }
  uint32 krem = sh_krem;
  uint32 running = ps[t] - local;  // exclusive prefix (in descending rank)
  for (int i = 0; i < CH; ++i) {
    int b = NB - 1 - (t * CH + i);
    uint32 c = hist[b];
    if (running < krem && running + c >= krem) {  // unique crossing point
      st->prefix = (sh_prefix << bits) | (uint32)b;
      st->k_rem = krem - running;
    }
    running += c;
  }
}

// Collect all candidates with ordered-score >= threshold T (count >= KSEL).
// Key packs (score, ~index) so a descending sort matches jax top_k ties.
__global__ void collect_kernel(const float* __restrict__ scores,
                               const TopKState* __restrict__ st,
                               u64* __restrict__ cands,
                               uint32* __restrict__ cnt, int N) {
  uint32 T = st->prefix;  // full 32-bit threshold after 3 scans
  int stride = gridDim.x * blockDim.x;
  for (int n = blockIdx.x * blockDim.x + threadIdx.x; n < N; n += stride) {
    uint32 s = f2ord(scores[n]);
    if (s >= T) {
      uint32 pos = atomicAdd(cnt, 1u);
      if (pos < CAND_CAP)
        cands[pos] = ((u64)s << 32) | (u64)(0xFFFFFFFFu - (uint32)n);
    }
  }
}

// Bitonic sort CAND_CAP u64 keys descending in one workgroup (64KB LDS),
// emit first KSEL sorted keys.
__global__ void sort_topk_kernel(const u64* __restrict__ cands,
                                 u64* __restrict__ topk) {
  __shared__ u64 sa[CAND_CAP];
  int t = threadIdx.x;
  for (int i = t; i < CAND_CAP; i += blockDim.x) sa[i] = cands[i];
  __syncthreads();
  for (unsigned k = 2; k <= CAND_CAP; k <<= 1) {
    for (unsigned j = k >> 1; j > 0; j >>= 1) {
      for (int i = t; i < CAND_CAP; i += blockDim.x) {
        unsigned ixj = (unsigned)i ^ j;
        if (ixj > (unsigned)i) {
          u64 x = sa[i], y = sa[ixj];
          bool desc = (((unsigned)i & k) == 0);
          if (desc ? (x < y) : (x > y)) { sa[i] = y; sa[ixj] = x; }
        }
      }
      __syncthreads();
    }
  }
  for (int i = t; i < KSEL; i += blockDim.x) topk[i] = sa[i];
}

// ---------------------------------------------------------------------------
// Phase 3: out[f, j] = node_embs[idx_j, f] * tanh(scores[idx_j])
// ---------------------------------------------------------------------------
__global__ void gather_out_kernel(const float* __restrict__ embs,
                                  const float* __restrict__ scores,
                                  const u64* __restrict__ topk,
                                  float* __restrict__ out, int N) {
  int j = blockIdx.x;
  u64 key = topk[j];
  uint32 n = 0xFFFFFFFFu - (uint32)(key & 0xFFFFFFFFull);
  if (n >= (uint32)N) n = 0;  // safety (only hit if padding leaked in)
  float w = tanhf(scores[n]);
  int f = threadIdx.x;
  out[(size_t)f * KSEL + j] = embs[(size_t)n * FDIM + f] * w;
}

// ---------------------------------------------------------------------------
extern "C" void kernel_launch(void* const* d_in, const int* in_sizes, int n_in,
                              void* d_out, int out_size, void* d_ws, size_t ws_size,
                              hipStream_t stream) {
  const float* node_embs = (const float*)d_in[0];
  const float* mask      = (const float*)d_in[1];
  const float* scorer    = (const float*)d_in[2];
  // d_in[3] = k (4096, compile-time KSEL)

  int N = in_sizes[1];  // mask has N elements

  // Workspace layout (256B aligned)
  char* ws = (char*)d_ws;
  size_t off = 0;
  auto take = [&](size_t bytes) {
    void* p = ws + off;
    off += (bytes + 255) & ~(size_t)255;
    return p;
  };
  float*      scores = (float*)take((size_t)N * 4);
  float*      sU     = (float*)take(FDIM * 4);
  uint32*     hist   = (uint32*)take(4096 * 4);
  TopKState*  st     = (TopKState*)take(sizeof(TopKState));
  uint32*     cnt    = (uint32*)take(4);
  u64*        cands  = (u64*)take(CAND_CAP * 8);
  u64*        topk   = (u64*)take(KSEL * 8);
  (void)ws_size;

  // Phase 0: unit scorer
  prep_scorer_kernel<<<1, FDIM, 0, stream>>>(scorer, sU);

  // Phase 1: scores via WMMA (16 rows per wave, 8 waves per block)
  int score_blocks = (N + 127) / 128;
  score_wmma_kernel<<<score_blocks, 256, 0, stream>>>(node_embs, sU, mask,
                                                      scores, N);

  // Top-k state init
  init_misc_kernel<<<(CAND_CAP + 255) / 256, 256, 0, stream>>>(st, cnt, cands);

  // 3-level radix select: 12 + 12 + 8 bits
  const int NBs[3]  = {4096, 4096, 256};
  const int bits[3] = {12, 12, 8};
  for (int level = 0; level < 3; ++level) {
    zero_hist_kernel<<<16, 256, 0, stream>>>(hist);
    hist_pass_kernel<<<1024, 256, 0, stream>>>(scores, hist, st, N, level);
    scan_pass_kernel<<<1, 256, 0, stream>>>(hist, st, NBs[level], bits[level]);
  }

  // Collect candidates >= threshold, sort descending (value, then index)
  collect_kernel<<<1024, 256, 0, stream>>>(scores, st, cands, cnt, N);
  sort_topk_kernel<<<1, 1024, 0, stream>>>(cands, topk);

  // Gather + weight + transpose
  gather_out_kernel<<<KSEL, FDIM, 0, stream>>>(node_embs, scores, topk,
                                               (float*)d_out, N);
}